// ICR_72327249264878
// MI455X (gfx1250) — compile-verified
//
#include <hip/hip_runtime.h>
#include <math.h>

typedef __attribute__((ext_vector_type(2))) float v2f;
typedef __attribute__((ext_vector_type(8))) float v8f;

#define C_IN   500
#define PP     2048
#define EPS_F  1e-7f
#define IOU_T  0.5f

// ---------------- init: zero the two global accumulators ----------------
__global__ void icr_init(float* __restrict__ sums) {
  sums[0] = 0.0f;
  sums[1] = 0.0f;
}

// ---------------- kernel 1: fc (WMMA f32 16x16x4) + softmax -> xr_k ----------------
// block = 128 threads = 4 waves; each wave computes 16 output rows.
__global__ __launch_bounds__(128) void icr_gemm_softmax(
    const float* __restrict__ inputs,   // (nrow, 500)
    const float* __restrict__ fc_w,     // (4, 500)
    const float* __restrict__ fc_b,     // (4,)
    float* __restrict__ xr)             // (nrow, 4) -> d_out
{
  const int wave = threadIdx.x >> 5;
  const int lane = threadIdx.x & 31;
  const int half = lane >> 4;          // 0: K{k,k+1}, 1: K{k+2,k+3}
  const int l16  = lane & 15;
  const long row0 = ((long)blockIdx.x * 4 + wave) * 16;

  // A fragment stream: lane reads cols {4t+2*half, 4t+2*half+1} of its row.
  const float2* ap2 = (const float2*)(inputs + (row0 + l16) * C_IN) + half;
  // B fragment stream: lanes with l16>=4 read row 0 (valid memory), zeroed below.
  const bool bvalid = (l16 < 4);
  const float2* bp2 =
      (const float2*)(fc_w + (long)(bvalid ? l16 : 0) * C_IN) + half;

  v8f acc = {};
  #pragma unroll 5
  for (int t = 0; t < C_IN / 4; ++t) {          // 125 WMMA steps, K = 500 exactly
    float2 af = ap2[2 * t];
    float2 bl = bp2[2 * t];
    v2f a;  a.x = af.x;  a.y = af.y;            // A: 16x4, M=l16, K=k0,k0+1
    v2f bb;                                     // B: 4x16, N=l16 (cols 4..15 zero)
    bb.x = bvalid ? bl.x : 0.0f;
    bb.y = bvalid ? bl.y : 0.0f;
    acc = __builtin_amdgcn_wmma_f32_16x16x4_f32(
        false, a, false, bb, (short)0, acc, false, false);
  }

  // D layout: lanes 0-15 -> (M=r, N=lane); lanes 16-31 -> (M=8+r, N=lane-16)
  __shared__ float tile[4][16][17];
  #pragma unroll
  for (int r = 0; r < 8; ++r)
    tile[wave][half * 8 + r][l16] = acc[r];
  __syncthreads();

  // 64 rows per block: threads 0..63 do bias + softmax over the 4 real columns
  if (threadIdx.x < 64) {
    const int w2 = threadIdx.x >> 4;
    const int m  = threadIdx.x & 15;
    float v0 = tile[w2][m][0] + fc_b[0];
    float v1 = tile[w2][m][1] + fc_b[1];
    float v2 = tile[w2][m][2] + fc_b[2];
    float v3 = tile[w2][m][3] + fc_b[3];
    float mx = fmaxf(fmaxf(v0, v1), fmaxf(v2, v3));
    float e0 = __expf(v0 - mx), e1 = __expf(v1 - mx);
    float e2 = __expf(v2 - mx), e3 = __expf(v3 - mx);
    float inv = 1.0f / (e0 + e1 + e2 + e3);
    const long orow = ((long)blockIdx.x * 4 + w2) * 16 + m;
    float4 o; o.x = e0 * inv; o.y = e1 * inv; o.z = e2 * inv; o.w = e3 * inv;
    *(float4*)(xr + orow * 4) = o;
  }
}

// ---------------- kernel 2: candidate count + first-argmax per (b,c) ----------------
__global__ __launch_bounds__(256) void icr_cand(
    const float* __restrict__ pre_score,   // (bs, P, 3)
    int* __restrict__ ncand, int* __restrict__ jmax)
{
  const int b = blockIdx.x / 3, c = blockIdx.x % 3;
  const float* ps = pre_score + (long)b * PP * 3 + c;
  int cnt = 0; float best = -1.0f; int bidx = 0;
  for (int j = threadIdx.x; j < PP; j += 256) {
    float v = ps[(long)j * 3];
    cnt += (v > 0.5f) ? 1 : 0;
    if (v > best) { best = v; bidx = j; }   // strict > keeps first max in-thread
  }
  __shared__ int   scnt[256];
  __shared__ float sbst[256];
  __shared__ int   sidx[256];
  scnt[threadIdx.x] = cnt; sbst[threadIdx.x] = best; sidx[threadIdx.x] = bidx;
  __syncthreads();
  for (int off = 128; off > 0; off >>= 1) {
    if (threadIdx.x < off) {
      scnt[threadIdx.x] += scnt[threadIdx.x + off];
      float ob = sbst[threadIdx.x + off]; int oi = sidx[threadIdx.x + off];
      if (ob > sbst[threadIdx.x] ||
          (ob == sbst[threadIdx.x] && oi < sidx[threadIdx.x])) {
        sbst[threadIdx.x] = ob; sidx[threadIdx.x] = oi;
      }
    }
    __syncthreads();
  }
  if (threadIdx.x == 0) { ncand[blockIdx.x] = scnt[0]; jmax[blockIdx.x] = sidx[0]; }
}

// ---------------- kernel 3: IoU scan + class resolution + focal loss ----------------
// grid = (bs, P/256); block = 256; one row i per thread, boxes staged in LDS.
__global__ __launch_bounds__(256) void icr_assign(
    const float* __restrict__ pre_score,     // (bs, P, 3)
    const unsigned char* __restrict__ labels,// (bs, 3) bool8
    const float* __restrict__ rois,          // (bs, P, 4)
    const float* __restrict__ xr,            // (bs*P, 4)
    const int* __restrict__ ncand, const int* __restrict__ jmax,
    float* __restrict__ sums)                // [0]=sum_w, [1]=sum_floss
{
  const int b = blockIdx.x;
  __shared__ float4        sbox[PP];
  __shared__ unsigned char scand[PP];

  const float4* rb = (const float4*)rois + (long)b * PP;
  const int nc0 = ncand[b*3+0], nc1 = ncand[b*3+1], nc2 = ncand[b*3+2];
  const int jm0 = jmax[b*3+0],  jm1 = jmax[b*3+1],  jm2 = jmax[b*3+2];
  for (int j = threadIdx.x; j < PP; j += 256) {
    sbox[j] = rb[j];
    const float* ps = pre_score + ((long)b * PP + j) * 3;
    unsigned char m = 0;
    m |= ((nc0 > 1) ? (ps[0] > 0.5f) : (j == jm0)) ? 1 : 0;
    m |= ((nc1 > 1) ? (ps[1] > 0.5f) : (j == jm1)) ? 2 : 0;
    m |= ((nc2 > 1) ? (ps[2] > 0.5f) : (j == jm2)) ? 4 : 0;
    scand[j] = m;
  }
  __syncthreads();

  const int i = blockIdx.y * 256 + threadIdx.x;
  const float4 bi = sbox[i];
  const float ai = (bi.z - bi.x) * (bi.w - bi.y);
  float best0 = -1.0f, best1 = -1.0f, best2 = -1.0f;
  int   j0 = 0, j1 = 0, j2 = 0;

  for (int j = 0; j < PP; ++j) {
    const unsigned char m = scand[j];      // uniform across block -> scalar skip
    if (!m) continue;
    const float4 bj = sbox[j];             // LDS broadcast
    float ltx = fmaxf(bi.x, bj.x), lty = fmaxf(bi.y, bj.y);
    float rbx = fminf(bi.z, bj.z), rby = fminf(bi.w, bj.w);
    float w = fmaxf(rbx - ltx, 0.0f), h = fmaxf(rby - lty, 0.0f);
    float inter = w * h;
    float aj = (bj.z - bj.x) * (bj.w - bj.y);
    float iou = inter / (ai + aj - inter);
    if ((m & 1) && iou > best0) { best0 = iou; j0 = j; }   // first-max tie-break
    if ((m & 2) && iou > best1) { best1 = iou; j1 = j; }
    if ((m & 4) && iou > best2) { best2 = iou; j2 = j; }
  }

  // sequential c=0,1,2 strict-> resolution (earlier class wins ties)
  float I = 0.0f, wv = 0.0f; int target = 3;
  {
    float bst = labels[b*3+0] ? best0 : -1.0f;
    if (bst > I) { wv = pre_score[((long)b*PP + j0)*3 + 0];
                   if (bst > IOU_T && target == 3) target = 0; I = bst; }
    bst = labels[b*3+1] ? best1 : -1.0f;
    if (bst > I) { wv = pre_score[((long)b*PP + j1)*3 + 1];
                   if (bst > IOU_T && target == 3) target = 1; I = bst; }
    bst = labels[b*3+2] ? best2 : -1.0f;
    if (bst > I) { wv = pre_score[((long)b*PP + j2)*3 + 2];
                   if (bst > IOU_T && target == 3) target = 2; I = bst; }
  }

  // focal loss on double-softmaxed scores
  const long n = (long)b * PP + i;
  float x0 = xr[n*4+0], x1 = xr[n*4+1], x2 = xr[n*4+2], x3 = xr[n*4+3];
  float mx = fmaxf(fmaxf(x0, x1), fmaxf(x2, x3));
  float e0 = __expf(x0 - mx), e1 = __expf(x1 - mx);
  float e2 = __expf(x2 - mx), e3 = __expf(x3 - mx);
  float s = e0 + e1 + e2 + e3;
  float pt = ((target == 0) ? e0 : (target == 1) ? e1 : (target == 2) ? e2 : e3) / s;
  pt = fminf(fmaxf(pt, EPS_F), 1.0f - EPS_F);
  float om = 1.0f - pt;
  float fl = -__logf(pt) * om * om;

  // block reductions + global atomics
  __shared__ float red[256];
  red[threadIdx.x] = wv;
  __syncthreads();
  for (int off = 128; off > 0; off >>= 1) {
    if (threadIdx.x < off) red[threadIdx.x] += red[threadIdx.x + off];
    __syncthreads();
  }
  if (threadIdx.x == 0) atomicAdd(&sums[0], red[0]);
  __syncthreads();
  red[threadIdx.x] = fl;
  __syncthreads();
  for (int off = 128; off > 0; off >>= 1) {
    if (threadIdx.x < off) red[threadIdx.x] += red[threadIdx.x + off];
    __syncthreads();
  }
  if (threadIdx.x == 0) atomicAdd(&sums[1], red[0]);
}

// ---------------- kernel 4: loss = mean(w) * mean(floss) ----------------
__global__ void icr_finish(const float* __restrict__ sums,
                           float* __restrict__ out_loss, int nrow) {
  const float inv = 1.0f / (float)nrow;
  out_loss[0] = (sums[0] * inv) * (sums[1] * inv);
}

// ---------------- launch ----------------
extern "C" void kernel_launch(void* const* d_in, const int* in_sizes, int n_in,
                              void* d_out, int out_size, void* d_ws, size_t ws_size,
                              hipStream_t stream) {
  const float*         inputs    = (const float*)d_in[0];
  const float*         pre_score = (const float*)d_in[1];
  const unsigned char* labels    = (const unsigned char*)d_in[2]; // jnp bool -> 1B
  const float*         rois      = (const float*)d_in[3];
  // d_in[4] = num (2048), fixed by construction
  const float*         fc_w      = (const float*)d_in[5];
  const float*         fc_b      = (const float*)d_in[6];
  float* out = (float*)d_out;

  const int nrow = in_sizes[0] / C_IN;   // bs * P = 32768
  const int bs   = nrow / PP;            // 16

  float* sums  = (float*)d_ws;           // 2 floats
  int*   ncand = (int*)d_ws + 8;         // 3*bs ints
  int*   jmax  = (int*)d_ws + 8 + 3 * bs;

  icr_init<<<1, 1, 0, stream>>>(sums);
  icr_gemm_softmax<<<nrow / 64, 128, 0, stream>>>(inputs, fc_w, fc_b, out);
  icr_cand<<<bs * 3, 256, 0, stream>>>(pre_score, ncand, jmax);
  icr_assign<<<dim3(bs, PP / 256), 256, 0, stream>>>(pre_score, labels, rois, out,
                                                     ncand, jmax, sums);
  icr_finish<<<1, 1, 0, stream>>>(sums, out + (long)nrow * 4, nrow);
}